// DeepJ_20237885898889
// MI455X (gfx1250) — compile-verified
//
#include <hip/hip_runtime.h>
#include <math.h>

typedef float v2f __attribute__((ext_vector_type(2)));
typedef float v8f __attribute__((ext_vector_type(8)));

#define NA   388   // NUM_ACTIONS
#define NS   8     // NUM_STYLES
#define SU   32
#define U    512
#define BB   16    // batch
#define SS   512   // seq
#define WIN  128
#define NWG  16    // RNN workgroups; each owns U/NWG = 32 columns
#define COLS 32

// workspace layout (float offsets)
#define OFF_HIN   0
#define OFF_HSEQ  (BB*SS*U)                 // 4,194,304
#define OFF_STB   (2*BB*SS*U)               // 8,388,608
#define OFF_ATT   (OFF_STB + BB*U)          // + 8192
#define OFF_LN    (OFF_ATT + NWG*BB*WIN)    // + 32768
#define OFF_BAR   (OFF_LN  + NWG*BB*2)      // + 512

// async global->LDS copy of 16 bytes (CDNA5; ASYNCcnt-tracked, no VGPR bounce)
__device__ __forceinline__ void async_copy_b128(unsigned lds_off,
                                                const float* gaddr) {
    unsigned long long ga = (unsigned long long)(const void*)gaddr;
    asm volatile("global_load_async_to_lds_b128 %0, %1, off"
                 :: "v"(lds_off), "v"(ga) : "memory");
}
__device__ __forceinline__ void wait_async0() {
    asm volatile("s_wait_asynccnt 0" ::: "memory");
}

// ---------------------------------------------------------------- style bias
// stb[b,u] = in_b[u] + sum_su (style[b]@style_w + style_b)[su] * in_w[388+su,u]
__global__ void k_style_bias(const float* __restrict__ style,
                             const float* __restrict__ style_w,
                             const float* __restrict__ style_b,
                             const float* __restrict__ in_w,
                             const float* __restrict__ in_b,
                             float* __restrict__ stb) {
    int id = blockIdx.x * blockDim.x + threadIdx.x;
    if (id >= BB * U) return;
    int b = id / U, u = id % U;
    float acc = in_b[u];
    for (int su = 0; su < SU; ++su) {
        float st = style_b[su];
        for (int k = 0; k < NS; ++k) st += style[b * NS + k] * style_w[k * SU + su];
        acc += st * in_w[(size_t)(NA + su) * U + u];
    }
    stb[id] = acc;
}

// ---------------------------------------------------------------- input GEMM
// hin[r,u] = relu( x[r,:388] @ in_w[:388,:] + stb[b(r),u] ),  r = b*S+s  (8192 rows)
__global__ void k_in_gemm(const float* __restrict__ x,
                          const float* __restrict__ in_w,
                          const float* __restrict__ stb,
                          float* __restrict__ hin) {
    int wv = threadIdx.x >> 5, lane = threadIdx.x & 31;
    int tile = blockIdx.x * 8 + wv;      // 512 mtiles x 32 ntiles = 16384 tiles
    int mt = tile >> 5, nt = tile & 31;
    int r0 = mt * 16;
    int l15 = lane & 15, hi = lane >> 4;
    int n = nt * 16 + l15;
    v8f acc = {0.f,0.f,0.f,0.f,0.f,0.f,0.f,0.f};
    const float* xrow = x + (size_t)(r0 + l15) * NA;   // A row m = lane%16
    for (int k = 0; k < NA; k += 4) {                  // 388 = 4*97, no tail
        int ka = k + hi * 2;
        v2f a, bf;
        a[0]  = xrow[ka];
        a[1]  = xrow[ka + 1];
        bf[0] = in_w[(size_t)ka * U + n];
        bf[1] = in_w[(size_t)(ka + 1) * U + n];
        acc = __builtin_amdgcn_wmma_f32_16x16x4_f32(false, a, false, bf,
                                                    (short)0, acc, false, false);
    }
    for (int i = 0; i < 8; ++i) {
        int r = r0 + i + hi * 8;         // C VGPR i: rows i (lo half) / i+8 (hi half)
        int bidx = r >> 9;               // r / S
        float v = acc[i] + stb[bidx * U + n];
        hin[(size_t)r * U + n] = v > 0.f ? v : 0.f;
    }
}

// ---------------------------------------------------------------- grid barrier
__device__ __forceinline__ void grid_barrier(unsigned* bar) {
    __syncthreads();
    if (threadIdx.x == 0) {
        __threadfence();
        unsigned gen = __hip_atomic_load(&bar[1], __ATOMIC_RELAXED, __HIP_MEMORY_SCOPE_AGENT);
        unsigned arrived = __hip_atomic_fetch_add(&bar[0], 1u, __ATOMIC_ACQ_REL,
                                                  __HIP_MEMORY_SCOPE_AGENT);
        if (arrived == (unsigned)gridDim.x - 1u) {
            __hip_atomic_store(&bar[0], 0u, __ATOMIC_RELAXED, __HIP_MEMORY_SCOPE_AGENT);
            __hip_atomic_fetch_add(&bar[1], 1u, __ATOMIC_RELEASE, __HIP_MEMORY_SCOPE_AGENT);
        } else {
            while (__hip_atomic_load(&bar[1], __ATOMIC_ACQUIRE,
                                     __HIP_MEMORY_SCOPE_AGENT) == gen)
                __builtin_amdgcn_s_sleep(2);
        }
        __threadfence();
    }
    __syncthreads();
}

// ---------------------------------------------------------------- RNN scan
// 16 persistent WGs, 128 threads each; WG g owns columns [g*32, g*32+32).
__global__ void k_rnn(const float* __restrict__ hin, float* __restrict__ hseq,
                      const float* __restrict__ rnn_w, const float* __restrict__ rnn_b,
                      const float* __restrict__ gamma, const float* __restrict__ beta,
                      float* __restrict__ att_part, float* __restrict__ ln_part,
                      unsigned* __restrict__ bar) {
    __shared__ float A_lds[16 * 260];       // padded rows: bank-conflict-free A frags
    __shared__ float pre_lds[BB * COLS];
    __shared__ float h1_lds[BB * COLS];
    __shared__ float attd_lds[BB * WIN];
    __shared__ float v_lds[BB * COLS];
    __shared__ float decay_lds[WIN];
    __shared__ float lnsum[BB], lnsum2[BB];
    __shared__ float mean_s[BB], std_s[BB];

    const int g   = blockIdx.x;
    const int tid = threadIdx.x;
    const int wv  = tid >> 5, lane = tid & 31;
    const int l15 = lane & 15, hi = lane >> 4;
    const int c0  = g * COLS;

    decay_lds[tid & (WIN - 1)] = 0.5f * powf(0.95f, (float)((tid & (WIN - 1)) + 1));
    __syncthreads();

    for (int t = 0; t < SS; ++t) {
        // ---- stage 1: preact = [x_t | h_{t-1}] @ rnn_w[:, cols] + rnn_b ----
        v8f acc = {0.f,0.f,0.f,0.f,0.f,0.f,0.f,0.f};
        for (int ch = 0; ch < 4; ++ch) {            // K = 1024 in 4 chunks of 256
            __syncthreads();
            for (int i = 0; i < 8; ++i) {           // stage A chunk into LDS (async)
                int f  = (tid + i * 128) * 4;       // 0..4095
                int bb = f >> 8, kk = f & 255;
                int kg = ch * 256 + kk;
                unsigned loff = (unsigned)(unsigned long long)(uintptr_t)
                                &A_lds[bb * 260 + kk];
                if (kg < U) {
                    async_copy_b128(loff, hin + ((size_t)(bb * SS + t)) * U + kg);
                } else if (t > 0) {
                    async_copy_b128(loff,
                        hseq + ((size_t)(bb * SS + (t - 1))) * U + (kg - U));
                } else {
                    *reinterpret_cast<float4*>(&A_lds[bb * 260 + kk]) =
                        make_float4(0.f, 0.f, 0.f, 0.f);
                }
            }
            wait_async0();                          // ASYNCcnt == 0: LDS tile ready
            __syncthreads();
            if (wv < 2) {                           // waves 0,1: one N-tile each
                int col = c0 + wv * 16 + l15;
                if (ch < 3) {                       // prefetch next rnn_w K-chunk
                    for (int pk = 0; pk < 256; pk += 64)
                        __builtin_prefetch(
                            &rnn_w[(size_t)((ch + 1) * 256 + pk) * U + col], 0, 1);
                }
                for (int kk = 0; kk < 256; kk += 4) {
                    int ko = kk + hi * 2;
                    int kg = ch * 256 + ko;
                    v2f a, bf;
                    a[0]  = A_lds[l15 * 260 + ko];
                    a[1]  = A_lds[l15 * 260 + ko + 1];
                    bf[0] = rnn_w[(size_t)kg * U + col];
                    bf[1] = rnn_w[(size_t)(kg + 1) * U + col];
                    acc = __builtin_amdgcn_wmma_f32_16x16x4_f32(
                              false, a, false, bf, (short)0, acc, false, false);
                }
            }
        }
        if (wv < 2) {
            int cl = wv * 16 + l15;
            for (int i = 0; i < 8; ++i) {
                int m = i + hi * 8;                 // m = batch index
                float v = acc[i] + rnn_b[c0 + cl];
                pre_lds[m * COLS + cl] = v;
                h1_lds[m * COLS + cl]  = v > 0.f ? v : 0.f;
            }
        }
        __syncthreads();

        // ---- stage 2: partial attention scores over this WG's 32 columns ----
        {
            int b = tid >> 3, a0 = tid & 7;
            for (int a = a0; a < WIN; a += 8) {
                int tt = t - 1 - a;
                float s = 0.f;
                if (tt >= 0) {
                    const float* hr = hseq + ((size_t)(b * SS + tt)) * U + c0;
                    for (int c = 0; c < COLS; ++c) s += hr[c] * h1_lds[b * COLS + c];
                }
                att_part[((size_t)g * BB + b) * WIN + a] = s;
            }
        }
        grid_barrier(bar);

        // ---- stage 3a: full att, fold 0.5*decay^(a+1) ----
        if (tid < BB) { lnsum[tid] = 0.f; lnsum2[tid] = 0.f; }
        for (int j = 0; j < 16; ++j) {
            int idx = tid + j * 128;                // 0..2047
            int b = idx >> 7, a = idx & 127;
            float s = 0.f;
            for (int gg = 0; gg < NWG; ++gg)
                s += att_part[((size_t)gg * BB + b) * WIN + a];
            attd_lds[b * WIN + a] = s * decay_lds[a];
        }
        __syncthreads();

        // ---- stage 3b: a_h + v, partial LN stats ----
        int amax = t < WIN ? t : WIN;
        for (int j = 0; j < 4; ++j) {
            int idx = tid + j * 128;                // 0..511
            int b = idx >> 5, c = idx & 31;
            float ah = 0.f;
            for (int a = 0; a < amax; ++a)
                ah += attd_lds[b * WIN + a] *
                      hseq[((size_t)(b * SS + (t - 1 - a))) * U + c0 + c];
            float v = pre_lds[b * COLS + c] + ah;
            v_lds[b * COLS + c] = v;
            atomicAdd(&lnsum[b],  v);
            atomicAdd(&lnsum2[b], v * v);
        }
        __syncthreads();
        if (tid < BB) {
            ln_part[((size_t)g * BB + tid) * 2 + 0] = lnsum[tid];
            ln_part[((size_t)g * BB + tid) * 2 + 1] = lnsum2[tid];
        }
        grid_barrier(bar);

        // ---- stage 4: LN + relu, emit h_t ----
        if (tid < BB) {
            float s1 = 0.f, s2 = 0.f;
            for (int gg = 0; gg < NWG; ++gg) {
                s1 += ln_part[((size_t)gg * BB + tid) * 2 + 0];
                s2 += ln_part[((size_t)gg * BB + tid) * 2 + 1];
            }
            float m   = s1 / (float)U;
            float var = s2 / (float)U - m * m;
            mean_s[tid] = m;
            std_s[tid]  = sqrtf(var > 0.f ? var : 0.f);
        }
        __syncthreads();
        for (int j = 0; j < 4; ++j) {
            int idx = tid + j * 128;
            int b = idx >> 5, c = idx & 31;
            float v = v_lds[b * COLS + c];
            float h = gamma[c0 + c] * (v - mean_s[b]) / (std_s[b] + 1e-6f) + beta[c0 + c];
            h = h > 0.f ? h : 0.f;
            hseq[((size_t)(b * SS + t)) * U + c0 + c] = h;
        }
        grid_barrier(bar);
    }
}

// ---------------------------------------------------------------- output GEMM
// out[r,n] = hseq[r,:] @ out_w[:,n] + out_b[n]   (N=388 -> 25 tiles, tail-guarded)
__global__ void k_out_gemm(const float* __restrict__ hseq,
                           const float* __restrict__ out_w,
                           const float* __restrict__ out_b,
                           float* __restrict__ out) {
    int wv = threadIdx.x >> 5, lane = threadIdx.x & 31;
    int tile = blockIdx.x * 8 + wv;      // 512 x 25 = 12800 tiles
    int mt = tile / 25, nt = tile % 25;
    int r0 = mt * 16;
    int l15 = lane & 15, hi = lane >> 4;
    int n = nt * 16 + l15;
    bool nok = n < NA;
    v8f acc = {0.f,0.f,0.f,0.f,0.f,0.f,0.f,0.f};
    const float* arow = hseq + (size_t)(r0 + l15) * U;
    for (int k = 0; k < U; k += 4) {
        int ka = k + hi * 2;
        v2f a, bf;
        a[0]  = arow[ka];
        a[1]  = arow[ka + 1];
        bf[0] = nok ? out_w[(size_t)ka * NA + n] : 0.f;
        bf[1] = nok ? out_w[(size_t)(ka + 1) * NA + n] : 0.f;
        acc = __builtin_amdgcn_wmma_f32_16x16x4_f32(false, a, false, bf,
                                                    (short)0, acc, false, false);
    }
    if (nok) {
        float bias = out_b[n];
        for (int i = 0; i < 8; ++i) {
            int r = r0 + i + hi * 8;
            out[(size_t)r * NA + n] = acc[i] + bias;
        }
    }
}

// ---------------------------------------------------------------- launch
extern "C" void kernel_launch(void* const* d_in, const int* in_sizes, int n_in,
                              void* d_out, int out_size, void* d_ws, size_t ws_size,
                              hipStream_t stream) {
    (void)in_sizes; (void)n_in; (void)out_size; (void)ws_size;
    const float* x       = (const float*)d_in[0];
    const float* style   = (const float*)d_in[1];
    const float* style_w = (const float*)d_in[2];
    const float* style_b = (const float*)d_in[3];
    const float* in_w    = (const float*)d_in[4];
    const float* in_b    = (const float*)d_in[5];
    const float* rnn_w   = (const float*)d_in[6];
    const float* rnn_b   = (const float*)d_in[7];
    const float* gamma   = (const float*)d_in[8];
    const float* beta    = (const float*)d_in[9];
    const float* out_w   = (const float*)d_in[10];
    const float* out_b   = (const float*)d_in[11];

    float* ws       = (float*)d_ws;
    float* hin      = ws + OFF_HIN;
    float* hseq     = ws + OFF_HSEQ;
    float* stb      = ws + OFF_STB;
    float* att_part = ws + OFF_ATT;
    float* ln_part  = ws + OFF_LN;
    unsigned* bar   = (unsigned*)(ws + OFF_BAR);

    hipMemsetAsync(bar, 0, 2 * sizeof(unsigned), stream);
    k_style_bias<<<(BB * U + 255) / 256, 256, 0, stream>>>(style, style_w, style_b,
                                                           in_w, in_b, stb);
    k_in_gemm<<<2048, 256, 0, stream>>>(x, in_w, stb, hin);
    k_rnn<<<NWG, 128, 0, stream>>>(hin, hseq, rnn_w, rnn_b, gamma, beta,
                                   att_part, ln_part, bar);
    k_out_gemm<<<1600, 256, 0, stream>>>(hseq, out_w, out_b, (float*)d_out);
}